// SwitchTransformersLoadBalancer_15599321219672
// MI455X (gfx1250) — compile-verified
//
#include <hip/hip_runtime.h>
#include <math.h>

typedef __attribute__((ext_vector_type(4))) float v4f;

#define CH    8
#define HW    65536            // 256*256
#define NPIX  (64 * HW)        // 4,194,304 pixels
#define TPB   256
#define QPT   4                // float4-quads per thread
#define NBLK  ((NPIX / 4) / (TPB * QPT))   // 1024 blocks

// --- tiny reduction: bias[c] = log(mean(accumulator[:, c])) -----------------
__global__ __launch_bounds__(256) void snorm_bias_kernel(
    const float* __restrict__ acc,   // [256, 8]
    float* __restrict__ bias)        // [8]
{
    __shared__ float sums[CH];
    const int t = threadIdx.x;
    if (t < CH) sums[t] = 0.0f;
    __syncthreads();

    const float* row = acc + (size_t)t * CH;
    float v[CH];
#pragma unroll
    for (int c = 0; c < CH; ++c) v[c] = row[c];
#pragma unroll
    for (int c = 0; c < CH; ++c) atomicAdd(&sums[c], v[c]);
    __syncthreads();

    if (t < CH) bias[t] = logf(sums[t] * (1.0f / 256.0f));
}

// --- fused softmax/SwitchNorm/argmax/one-hot --------------------------------
// Double-buffered ASYNC global->LDS pipeline (gfx1250 copy engine, ASYNCcnt),
// compute from LDS, non-temporal b128 one-hot stores.
__global__ __launch_bounds__(TPB) void route_top1_kernel(
    const float* __restrict__ x,     // [64, 8, 256, 256]
    const float* __restrict__ bias,  // [8] = log(norm[c])
    float* __restrict__ out)         // [64, 8, 256, 256]
{
    __shared__ v4f stage[2][CH][TPB];      // 2 x 32 KB double buffer

    float b[CH];
#pragma unroll
    for (int c = 0; c < CH; ++c) b[c] = bias[c];

    const int    tid       = threadIdx.x;
    const size_t blockBase = (size_t)blockIdx.x * (TPB * QPT);

    // Issue 8 per-thread async b128 copies (one per expert plane) for tile `it`
    auto issue_tile = [&](int it, int buf) {
        const size_t q   = blockBase + (size_t)it * TPB + tid;  // quad index
        const size_t p   = q << 2;                              // pixel index
        const size_t img = p >> 16;
        const size_t hw  = p & (size_t)(HW - 1);
        const float* plane0 = x + img * (size_t)(CH * HW) + hw;
#pragma unroll
        for (int c = 0; c < CH; ++c) {
            // low 32 bits of a flat pointer to __shared__ == LDS byte offset
            unsigned lds_addr = (unsigned)(uintptr_t)&stage[buf][c][tid];
            unsigned long long ga =
                (unsigned long long)(plane0 + (size_t)c * HW);
            asm volatile("global_load_async_to_lds_b128 %0, %1, off"
                         :: "v"(lds_addr), "v"(ga)
                         : "memory");
        }
    };

    issue_tile(0, 0);

#pragma unroll
    for (int it = 0; it < QPT; ++it) {
        if (it + 1 < QPT) {
            issue_tile(it + 1, (it + 1) & 1);
            // 8 copies of tile it+1 outstanding; <=8 ==> tile it complete
            asm volatile("s_wait_asynccnt 8" ::: "memory");
        } else {
            asm volatile("s_wait_asynccnt 0" ::: "memory");
        }

        const int    buf = it & 1;
        const size_t q   = blockBase + (size_t)it * TPB + tid;
        const size_t p   = q << 2;
        const size_t img = p >> 16;
        const size_t hw  = p & (size_t)(HW - 1);

        // Read staged tile back (ds_load_b128, thread-private slots: no barrier)
        v4f v[CH];
#pragma unroll
        for (int c = 0; c < CH; ++c) v[c] = stage[buf][c][tid];

        // 4-lane argmax of (x_c - log(norm_c)); strict '>' = first-index ties
        float bx = v[0].x - b[0], by = v[0].y - b[0];
        float bz = v[0].z - b[0], bw = v[0].w - b[0];
        int ix = 0, iy = 0, iz = 0, iw = 0;
#pragma unroll
        for (int c = 1; c < CH; ++c) {
            const float tx = v[c].x - b[c];
            const float ty = v[c].y - b[c];
            const float tz = v[c].z - b[c];
            const float tw = v[c].w - b[c];
            if (tx > bx) { bx = tx; ix = c; }
            if (ty > by) { by = ty; iy = c; }
            if (tz > bz) { bz = tz; iz = c; }
            if (tw > bw) { bw = tw; iw = c; }
        }

        // 8 non-temporal b128 one-hot stores (stream-once output)
        float* oplane0 = out + img * (size_t)(CH * HW) + hw;
#pragma unroll
        for (int c = 0; c < CH; ++c) {
            v4f o;
            o.x = (ix == c) ? 1.0f : 0.0f;
            o.y = (iy == c) ? 1.0f : 0.0f;
            o.z = (iz == c) ? 1.0f : 0.0f;
            o.w = (iw == c) ? 1.0f : 0.0f;
            __builtin_nontemporal_store(o, (v4f*)(oplane0 + (size_t)c * HW));
        }
    }
}

extern "C" void kernel_launch(void* const* d_in, const int* in_sizes, int n_in,
                              void* d_out, int out_size, void* d_ws, size_t ws_size,
                              hipStream_t stream) {
    const float* x   = (const float*)d_in[0];   // [64, 8, 256, 256] fp32
    const float* acc = (const float*)d_in[1];   // [256, 8] fp32
    float* out  = (float*)d_out;                // [64, 8, 256, 256] fp32
    float* bias = (float*)d_ws;                 // 8 floats of scratch

    snorm_bias_kernel<<<1, 256, 0, stream>>>(acc, bias);
    route_top1_kernel<<<NBLK, TPB, 0, stream>>>(x, bias, out);
    (void)in_sizes; (void)n_in; (void)out_size; (void)ws_size;
}